// Ive_36197984370718
// MI455X (gfx1250) — compile-verified
//
#include <hip/hip_runtime.h>
#include <hip/hip_bf16.h>

// Native clang vector type so __builtin_nontemporal_load/store lower cleanly
// to global_load_b128 / global_store_b128 with th:TH_*_NT cache hints.
typedef float v4f __attribute__((ext_vector_type(4)));

// Exponentially scaled modified Bessel i1e(x) = exp(-|x|) * I1(x), with
// sign(x) odd symmetry. Abramowitz & Stegun 9.8.3 / 9.8.4 rational approx,
// exactly matching the JAX reference's coefficients and branch structure.
// Branchless: both branches evaluated, v_cndmask selects.
__device__ __forceinline__ float i1e_one(float x) {
    const float ax = __builtin_fabsf(x);          // free src modifier

    // ---- small branch: ax <= 3.75 ----
    // t = (ax/3.75)^2
    float t = ax * 0.26666666666666666f;
    t = t * t;
    float ps = __builtin_fmaf(0.00032411f, t, 0.00301532f);
    ps = __builtin_fmaf(ps, t, 0.02658733f);
    ps = __builtin_fmaf(ps, t, 0.15084934f);
    ps = __builtin_fmaf(ps, t, 0.51498869f);
    ps = __builtin_fmaf(ps, t, 0.87890594f);
    ps = __builtin_fmaf(ps, t, 0.5f);
    // exp(-ax): hardware v_exp_f32 (arg <= 3.75 here, error negligible)
    float e = __expf(-ax);
    float r_small = ax * ps * e;

    // ---- large branch: ax > 3.75 ----
    float axl = __builtin_fmaxf(ax, 3.75f);       // avoid rcp(0) in dead lane path
    float tl  = 3.75f * __builtin_amdgcn_rcpf(axl);   // v_rcp_f32, ~1 ulp
    float pl = __builtin_fmaf(-0.00420059f, tl, 0.01787654f);
    pl = __builtin_fmaf(pl, tl, -0.02895312f);
    pl = __builtin_fmaf(pl, tl,  0.02282967f);
    pl = __builtin_fmaf(pl, tl, -0.01031555f);
    pl = __builtin_fmaf(pl, tl,  0.00163801f);
    pl = __builtin_fmaf(pl, tl, -0.00362018f);
    pl = __builtin_fmaf(pl, tl, -0.03988024f);
    pl = __builtin_fmaf(pl, tl,  0.39894228f);
    float r_large = pl * __builtin_amdgcn_rsqf(axl);  // v_rsq_f32, ~1 ulp

    float r = (ax <= 3.75f) ? r_small : r_large;
    // sign(x) * r  (r >= 0 mathematically; x==0 -> r==0 either way)
    return __builtin_copysignf(r, x);
}

__device__ __forceinline__ v4f i1e_v4(v4f a) {
    v4f r;
    r.x = i1e_one(a.x);
    r.y = i1e_one(a.y);
    r.z = i1e_one(a.z);
    r.w = i1e_one(a.w);
    return r;
}

__global__ __launch_bounds__(256) void i1e_kernel(const float* __restrict__ in,
                                                  float* __restrict__ out,
                                                  long long n) {
    const long long n4  = n >> 2;                                   // whole float4s
    const long long T   = (long long)gridDim.x * blockDim.x;        // grid stride
    const long long tid = (long long)blockIdx.x * blockDim.x + threadIdx.x;

    const v4f* __restrict__ in4  = (const v4f*)in;
    v4f* __restrict__       out4 = (v4f*)out;

    // Main loop: 4 non-temporal b128 loads in flight per thread per iteration,
    // each load perfectly coalesced per-instruction (lane stride 16B).
    long long i = tid;
    for (; i + 3 * T < n4; i += 4 * T) {
        v4f a = __builtin_nontemporal_load(in4 + i);
        v4f b = __builtin_nontemporal_load(in4 + i + T);
        v4f c = __builtin_nontemporal_load(in4 + i + 2 * T);
        v4f d = __builtin_nontemporal_load(in4 + i + 3 * T);

        v4f ra = i1e_v4(a);
        v4f rb = i1e_v4(b);
        v4f rc = i1e_v4(c);
        v4f rd = i1e_v4(d);

        __builtin_nontemporal_store(ra, out4 + i);
        __builtin_nontemporal_store(rb, out4 + i + T);
        __builtin_nontemporal_store(rc, out4 + i + 2 * T);
        __builtin_nontemporal_store(rd, out4 + i + 3 * T);
    }
    // Leftover whole-float4s
    for (; i < n4; i += T) {
        v4f a = __builtin_nontemporal_load(in4 + i);
        __builtin_nontemporal_store(i1e_v4(a), out4 + i);
    }
    // Scalar tail (n % 4) — empty for this problem size but kept general.
    for (long long j = (n4 << 2) + tid; j < n; j += T) {
        out[j] = i1e_one(in[j]);
    }
}

extern "C" void kernel_launch(void* const* d_in, const int* in_sizes, int n_in,
                              void* d_out, int out_size, void* d_ws, size_t ws_size,
                              hipStream_t stream) {
    (void)n_in; (void)d_ws; (void)ws_size;
    const float* z = (const float*)d_in[0];
    float* out = (float*)d_out;
    const long long n = (long long)out_size;   // == in_sizes[0] == 16*4096*1024

    // 4096 blocks x 256 threads (8 wave32/block) = 32K waves: saturates HBM,
    // main loop runs 4 iterations at this problem size.
    const int TPB = 256;
    const int BLOCKS = 4096;
    i1e_kernel<<<BLOCKS, TPB, 0, stream>>>(z, out, n);
}